// GeometricPositionalFingerprinter_46523085750350
// MI455X (gfx1250) — compile-verified
//
#include <hip/hip_runtime.h>
#include <hip/hip_bf16.h>

typedef __attribute__((ext_vector_type(2))) float v2f;
typedef __attribute__((ext_vector_type(8))) float v8f;
typedef int v4i __attribute__((vector_size(16)));  // matches builtin param type

#define WAVES_PER_BLOCK 8
#define ROW_STRIDE 132            // 128 + 4 pad floats: conflict-free WMMA-feed reads
#define STAGE_F (5 * ROW_STRIDE)  // 660 floats per stage buffer
#define GST_F (32 * 15)           // 32 pentachora * 15 unique Gram entries

#if __has_builtin(__builtin_amdgcn_global_load_async_to_lds_b128)
#define USE_ASYNC_LDS 1
#endif

#define AS1 __attribute__((address_space(1)))
#define AS3 __attribute__((address_space(3)))

__global__ __launch_bounds__(32 * WAVES_PER_BLOCK)
void cantor_fingerprint_kernel(const float* __restrict__ penta,
                               float* __restrict__ out, int N) {
  __shared__ __align__(16) float stage[WAVES_PER_BLOCK][2][STAGE_F];
  __shared__ float gst[WAVES_PER_BLOCK][GST_F];

  const int lane = threadIdx.x & 31;
  const int w    = threadIdx.x >> 5;
  const int gw   = blockIdx.x * (blockDim.x >> 5) + w;
  const int nw   = gridDim.x * (blockDim.x >> 5);
  const int NB   = (N + 31) >> 5;

  // WMMA A/B operand read base (fp32 16x4 A layout; rows>=5 clamped -> only
  // G entries outside the 5x5 block see garbage, and we never read those).
  const int hi     = lane >> 4;
  const int rowc   = min(lane & 15, 4);
  const int abBase = rowc * ROW_STRIDE + hi * 2;

  for (int bi = gw; bi < NB; bi += nw) {
    const int base = bi << 5;

#ifdef USE_ASYNC_LDS
    // prime: async-copy pentachoron `base` straight into LDS buffer 0 (no VGPR data)
    {
      const float* s = penta + (size_t)base * 640;
#pragma unroll
      for (int j = 0; j < 5; ++j)
        __builtin_amdgcn_global_load_async_to_lds_b128(
            (AS1 v4i*)(s + j * 128 + lane * 4),
            (AS3 v4i*)&stage[w][0][j * ROW_STRIDE + lane * 4], 0, 0);
    }
#endif

#pragma unroll 1
    for (int i = 0; i < 32; ++i) {
      const int buf   = i & 1;
      const int p     = base + i;                       // wave-uniform
      const bool valid = (p < N);
      const bool pre   = (i < 31) && (base + i + 1 < N);

#ifdef USE_ASYNC_LDS
      if (pre) {  // prefetch next pentachoron into the other buffer
        const float* s = penta + (size_t)(base + i + 1) * 640;
#pragma unroll
        for (int j = 0; j < 5; ++j)
          __builtin_amdgcn_global_load_async_to_lds_b128(
              (AS1 v4i*)(s + j * 128 + lane * 4),
              (AS3 v4i*)&stage[w][buf ^ 1][j * ROW_STRIDE + lane * 4], 0, 0);
      }
#endif

      if (valid) {
#ifdef USE_ASYNC_LDS
        // async loads retire in order: <=5 outstanding leaves only the prefetch in flight
        if (pre) asm volatile("s_wait_asynccnt 5" ::: "memory");
        else     asm volatile("s_wait_asynccnt 0" ::: "memory");
#else
        {  // fallback: synchronous stage fill
          const float* s = penta + (size_t)p * 640;
#pragma unroll
          for (int j = 0; j < 5; ++j) {
            float4 rj = ((const float4*)(s + j * 128))[lane];
            ((float4*)&stage[w][buf][j * ROW_STRIDE])[lane] = rj;
          }
          asm volatile("s_wait_dscnt 0" ::: "memory");
        }
#endif
        v8f a0 = {0,0,0,0,0,0,0,0}, a1 = {0,0,0,0,0,0,0,0};
        v8f a2 = {0,0,0,0,0,0,0,0}, a3 = {0,0,0,0,0,0,0,0};
        // G = V * V^T via 32 K-chunks of 4; A and B share the same VGPR image.
#pragma unroll
        for (int k = 0; k < 32; k += 4) {
          v2f x0 = *(const v2f*)&stage[w][buf][abBase + 4 * (k + 0)];
          v2f x1 = *(const v2f*)&stage[w][buf][abBase + 4 * (k + 1)];
          v2f x2 = *(const v2f*)&stage[w][buf][abBase + 4 * (k + 2)];
          v2f x3 = *(const v2f*)&stage[w][buf][abBase + 4 * (k + 3)];
          a0 = __builtin_amdgcn_wmma_f32_16x16x4_f32(false, x0, false, x0, (short)0, a0, false, false);
          a1 = __builtin_amdgcn_wmma_f32_16x16x4_f32(false, x1, false, x1, (short)0, a1, false, false);
          a2 = __builtin_amdgcn_wmma_f32_16x16x4_f32(false, x2, false, x2, (short)0, a2, false, false);
          a3 = __builtin_amdgcn_wmma_f32_16x16x4_f32(false, x3, false, x3, (short)0, a3, false, false);
        }
        v8f g8 = (a0 + a1) + (a2 + a3);
        // D layout: VGPR m holds row M=m, lanes 0..15 hold N=lane.
        // Stash the 15 unique Gram entries: entry (m,n) m<=n at n*(n+1)/2+m.
        if (lane < 5) {
          const int triBase = i * 15 + (lane * (lane + 1)) / 2;
#pragma unroll
          for (int m = 0; m < 5; ++m)
            if (m <= lane) gst[w][triBase + m] = g8[m];
        }
      }
    }

    asm volatile("s_wait_dscnt 0" ::: "memory");

    // ---- phase 2: one pentachoron per lane (scalar tail is lane-parallel)
    const int p2 = base + lane;
    if (p2 < N) {
      float gg[5][5];
#pragma unroll
      for (int m = 0; m < 5; ++m)
#pragma unroll
        for (int n = 0; n < 5; ++n) {
          const int a = m < n ? m : n, b = m < n ? n : m;
          gg[m][n] = gst[w][lane * 15 + (b * (b + 1)) / 2 + a];
        }

      // squared distances from Gram
      float d2[5][5];
#pragma unroll
      for (int a = 0; a < 5; ++a) {
        d2[a][a] = 0.0f;
#pragma unroll
        for (int b = a + 1; b < 5; ++b) {
          float v = fmaxf(gg[a][a] + gg[b][b] - 2.0f * gg[a][b], 0.0f);
          d2[a][b] = v; d2[b][a] = v;
        }
      }

      // edge stats over 10 unique edges (ddof=1)
      float e[10];
      {
        int t = 0;
#pragma unroll
        for (int a = 0; a < 5; ++a)
#pragma unroll
          for (int b = a + 1; b < 5; ++b)
            e[t++] = sqrtf(d2[a][b]);
      }
      float mean_e = 0.0f;
#pragma unroll
      for (int t = 0; t < 10; ++t) mean_e += e[t];
      mean_e *= 0.1f;
      float var_e = 0.0f;
#pragma unroll
      for (int t = 0; t < 10; ++t) { float d = e[t] - mean_e; var_e += d * d; }
      const float std_e = sqrtf(var_e * (1.0f / 9.0f));

      // 6x6 Cayley-Menger determinant; rows 0,1 pre-swapped so pivot0 = 1
      float m[6][6];
      m[0][0] = 1.0f; m[0][1] = 0.0f;
#pragma unroll
      for (int j = 2; j < 6; ++j) m[0][j] = d2[0][j - 1];
      m[1][0] = 0.0f;
#pragma unroll
      for (int j = 1; j < 6; ++j) m[1][j] = 1.0f;
#pragma unroll
      for (int i2 = 2; i2 < 6; ++i2) {
        m[i2][0] = 1.0f;
#pragma unroll
        for (int j = 1; j < 6; ++j) m[i2][j] = d2[i2 - 1][j - 1];
      }
      float det = -1.0f;  // one row swap
#pragma unroll
      for (int k = 0; k < 6; ++k) {
        const float piv = m[k][k];
        det *= piv;
        const float inv = 1.0f / piv;
#pragma unroll
        for (int i2 = k + 1; i2 < 6; ++i2) {
          const float f = m[i2][k] * inv;
#pragma unroll
          for (int j = k + 1; j < 6; ++j) m[i2][j] -= f * m[k][j];
        }
      }
      const float vol = sqrtf(fmaxf(-det * (1.0f / 9216.0f), 0.0f));

      // vertex spread about centroid, all from Gram (ddof=1)
      float rs[5], tot = 0.0f;
#pragma unroll
      for (int v = 0; v < 5; ++v) {
        float s = 0.0f;
#pragma unroll
        for (int n = 0; n < 5; ++n) s += gg[v][n];
        rs[v] = s; tot += s;
      }
      float cd[5], mean_c = 0.0f;
#pragma unroll
      for (int v = 0; v < 5; ++v) {
        cd[v] = sqrtf(fmaxf(gg[v][v] - 0.4f * rs[v] + 0.04f * tot, 0.0f));
        mean_c += cd[v];
      }
      mean_c *= 0.2f;
      float var_c = 0.0f;
#pragma unroll
      for (int v = 0; v < 5; ++v) { float d = cd[v] - mean_c; var_c += d * d; }
      const float spread = sqrtf(var_c * 0.25f);

      auto sig = [](float x) { return 1.0f / (1.0f + expf(-x)); };
      const float vn = sig(vol * 10.0f);
      const float er = sig(std_e / (mean_e + 1e-6f));
      const float sn = sig(spread);

      float x = fminf(fmaxf(0.4f * vn + 0.3f * er + 0.3f * sn, 1e-6f), 1.0f - 1e-6f);
      const float drift = (vn + er + sn) * 0.01f;
      float cv = 0.0f, fac = 0.5f;
#pragma unroll
      for (int t = 0; t < 8; ++t) {
        const float xs = x * 3.0f;
        const float dg = floorf(xs);
        const float fr = xs - dg;
        cv += (dg == 2.0f) ? fac : 0.0f;
        x = fminf(fmaxf(fr + drift, 1e-6f), 1.0f - 1e-6f);
        fac *= 0.5f;
      }
      out[p2] = fminf(fmaxf(cv, 0.0f), 1.0f);
    }
  }
}

extern "C" void kernel_launch(void* const* d_in, const int* in_sizes, int n_in,
                              void* d_out, int out_size, void* d_ws, size_t ws_size,
                              hipStream_t stream) {
  const float* penta = (const float*)d_in[0];
  float* out = (float*)d_out;
  const int N  = in_sizes[0] / 640;                   // [N,5,128] fp32
  const int NB = (N + 31) / 32;                       // 32 pentachora per wave-batch
  const int blocks = (NB + WAVES_PER_BLOCK - 1) / WAVES_PER_BLOCK;
  cantor_fingerprint_kernel<<<blocks, 32 * WAVES_PER_BLOCK, 0, stream>>>(penta, out, N);
}